// GCN_90881507983627
// MI455X (gfx1250) — compile-verified
//
#include <hip/hip_runtime.h>
#include <hip/hip_bf16.h>

// ---------------------------------------------------------------------------
// 2-layer GCN for MI455X (gfx1250, wave32).
//
// Roofline: X@W1 (1.64 GFLOP / 205 MB fp32) is memory-bound at 23.3 TB/s,
// so we use the exact-precision V_WMMA_F32_16X16X4_F32 path. f_hid == 16 is
// exactly one WMMA tile in N. Edge aggregation data (h1 = 6.4 MB) is
// L2-resident (192 MB L2), so gather + f32-atomic scatter stays on-chip.
// ---------------------------------------------------------------------------

typedef __attribute__((ext_vector_type(2))) float v2f;
typedef __attribute__((ext_vector_type(8))) float v8f;

#define F_IN  512
#define F_HID 16
#define F_OUT 7

// ---------------------------- degree / norm --------------------------------

__global__ __launch_bounds__(256) void k_init_deg(float* __restrict__ deg, int n) {
  int i = blockIdx.x * 256 + threadIdx.x;
  if (i < n) deg[i] = 1.0f;  // self-loop contributes 1
}

__global__ __launch_bounds__(256) void k_count_deg(const long long* __restrict__ ei,
                                                   float* __restrict__ deg, long long E) {
  long long e = (long long)blockIdx.x * 256 + threadIdx.x;
  if (e < E) atomicAdd(&deg[ei[E + e]], 1.0f);  // in-degree over dst
}

__global__ __launch_bounds__(256) void k_rsqrt_inplace(float* __restrict__ deg, int n) {
  int i = blockIdx.x * 256 + threadIdx.x;
  if (i < n) deg[i] = rsqrtf(deg[i]);  // deg >= 1 always (self-loops)
}

// ------------------------------ GEMM 1 -------------------------------------
// h1[N,16] = X[N,512] @ W1[512,16]; one wave32 per 16x16 tile, K-loop of
// 128 x V_WMMA_F32_16X16X4_F32. W1 staged in LDS (32 KB / block).

__global__ __launch_bounds__(256) void k_gemm1_wmma(const float* __restrict__ X,
                                                    const float* __restrict__ W1,
                                                    float* __restrict__ H1, int n_nodes) {
  __shared__ float sW[F_IN * F_HID];  // 32 KB of the 320 KB WGP LDS
  int tid = threadIdx.x;
  // Cooperative coalesced load of W1 into LDS (8192 floats / 256 threads).
  for (int i = tid; i < (F_IN * F_HID) / 4; i += 256)
    ((float4*)sW)[i] = ((const float4*)W1)[i];
  __syncthreads();

  int wave = tid >> 5;
  int lane = tid & 31;
  int hi = lane >> 4;   // half-wave select: K offset +2 (A/B), M offset +8 (C)
  int lo = lane & 15;   // M index for A, N index for B/C

  int row0 = (blockIdx.x * 8 + wave) * 16;
  if (row0 >= n_nodes) return;  // wave-uniform guard: EXEC all-1s at WMMA

  const float* xrow = X + (size_t)(row0 + lo) * F_IN + 2 * hi;  // A[lo][k+2*hi]
  const float* wcol = sW + 2 * hi * F_HID + lo;                 // B[k+2*hi][lo]

  v8f c = {0.f, 0.f, 0.f, 0.f, 0.f, 0.f, 0.f, 0.f};
#pragma unroll 4
  for (int k = 0; k < F_IN; k += 4) {
    v2f a = *(const v2f*)(xrow + k);          // global_load_b64: K=k..k+1 (+2*hi)
    v2f b;
    b.x = wcol[(k + 0) * F_HID];              // ds_load_b32
    b.y = wcol[(k + 1) * F_HID];
    c = __builtin_amdgcn_wmma_f32_16x16x4_f32(false, a, false, b, (short)0, c,
                                              false, false);
  }

  // C layout: lane(hi,lo), vgpr r -> element (M = r + 8*hi, N = lo).
  float* hp = H1 + (size_t)(row0 + 8 * hi) * F_HID + lo;
#pragma unroll
  for (int r = 0; r < 8; ++r) hp[r * F_HID] = c[r];  // coalesced 64B row chunks
}

// --------------------------- layer-1 aggregation ---------------------------

__global__ __launch_bounds__(256) void k_agg1_init(const float* __restrict__ h1,
                                                   const float* __restrict__ dinv,
                                                   const float* __restrict__ b1,
                                                   float* __restrict__ out1, int n_nodes) {
  int gid = blockIdx.x * 256 + threadIdx.x;
  if (gid >= n_nodes * F_HID) return;
  int i = gid >> 4, f = gid & 15;
  float di = dinv[i];
  out1[gid] = h1[gid] * di * di + b1[f];  // self-loop message + bias
}

__global__ __launch_bounds__(256) void k_agg1_edge(const long long* __restrict__ ei,
                                                   const float* __restrict__ h1,
                                                   const float* __restrict__ dinv,
                                                   float* __restrict__ out1, long long E) {
  long long gid = (long long)blockIdx.x * 256 + threadIdx.x;
  if (gid >= E * F_HID) return;
  long long e = gid >> 4;
  int f = (int)(gid & 15);
  long long s = ei[e], d = ei[E + e];
  float norm = dinv[s] * dinv[d];
  atomicAdd(&out1[d * F_HID + f], h1[s * F_HID + f] * norm);  // L2-resident
}

// ------------------------------ GEMM 2 -------------------------------------
// h2[N,8(pad)] = relu(out1)[N,16] @ W2[16,7]; ReLU fused into A reads,
// W2 zero-padded N=7 -> 16 via lane mask. K=16 -> 4 WMMA steps.

__global__ __launch_bounds__(256) void k_gemm2_wmma(const float* __restrict__ OUT1,
                                                    const float* __restrict__ W2,
                                                    float* __restrict__ H2, int n_nodes) {
  int tid = threadIdx.x;
  int wave = tid >> 5;
  int lane = tid & 31;
  int hi = lane >> 4, lo = lane & 15;

  int row0 = (blockIdx.x * 8 + wave) * 16;
  if (row0 >= n_nodes) return;  // wave-uniform

  float wmask = (lo < F_OUT) ? 1.0f : 0.0f;  // zero-pad B columns 7..15
  int wc = (lo < F_OUT) ? lo : (F_OUT - 1);  // clamped (non-divergent) load

  const float* arow = OUT1 + (size_t)(row0 + lo) * F_HID + 2 * hi;

  v8f c = {0.f, 0.f, 0.f, 0.f, 0.f, 0.f, 0.f, 0.f};
#pragma unroll
  for (int k = 0; k < F_HID; k += 4) {
    v2f a, b;
    a.x = fmaxf(arow[k + 0], 0.0f);  // fused ReLU
    a.y = fmaxf(arow[k + 1], 0.0f);
    b.x = W2[(k + 0 + 2 * hi) * F_OUT + wc] * wmask;
    b.y = W2[(k + 1 + 2 * hi) * F_OUT + wc] * wmask;
    c = __builtin_amdgcn_wmma_f32_16x16x4_f32(false, a, false, b, (short)0, c,
                                              false, false);
  }

  if (lo < 8) {  // only columns 0..7 exist in stride-8 h2 (col 7 never read)
    float* hp = H2 + (size_t)(row0 + 8 * hi) * 8 + lo;
#pragma unroll
    for (int r = 0; r < 8; ++r) hp[r * 8] = c[r];
  }
}

// --------------------------- layer-2 aggregation ---------------------------

__global__ __launch_bounds__(256) void k_agg2_init(const float* __restrict__ h2,
                                                   const float* __restrict__ dinv,
                                                   const float* __restrict__ b2,
                                                   float* __restrict__ out, int n_nodes) {
  int gid = blockIdx.x * 256 + threadIdx.x;
  if (gid >= n_nodes * 8) return;
  int i = gid >> 3, f = gid & 7;
  if (f >= F_OUT) return;
  float di = dinv[i];
  out[i * F_OUT + f] = h2[i * 8 + f] * di * di + b2[f];
}

__global__ __launch_bounds__(256) void k_agg2_edge(const long long* __restrict__ ei,
                                                   const float* __restrict__ h2,
                                                   const float* __restrict__ dinv,
                                                   float* __restrict__ out, long long E) {
  long long gid = (long long)blockIdx.x * 256 + threadIdx.x;
  if (gid >= E * 8) return;
  long long e = gid >> 3;
  int f = (int)(gid & 7);
  if (f >= F_OUT) return;
  long long s = ei[e], d = ei[E + e];
  float norm = dinv[s] * dinv[d];
  atomicAdd(&out[d * F_OUT + f], h2[s * 8 + f] * norm);
}

// ------------------------------- launcher ----------------------------------

extern "C" void kernel_launch(void* const* d_in, const int* in_sizes, int n_in,
                              void* d_out, int out_size, void* d_ws, size_t ws_size,
                              hipStream_t stream) {
  const float*     X   = (const float*)d_in[0];
  const long long* ei  = (const long long*)d_in[1];  // int64 per reference
  const float*     W1  = (const float*)d_in[2];
  const float*     b1  = (const float*)d_in[3];
  const float*     W2  = (const float*)d_in[4];
  const float*     b2  = (const float*)d_in[5];
  float*           out = (float*)d_out;

  const int       N = in_sizes[0] / F_IN;          // 100000
  const long long E = (long long)in_sizes[1] / 2;  // 3200000

  // Workspace carve-out (1 KB aligned): dinv | h1 | out1 | h2  (~16.4 MB)
  char* ws = (char*)d_ws;
  size_t off = 0;
  auto carve = [&](size_t bytes) {
    float* p = (float*)(ws + off);
    off += (bytes + 1023) & ~(size_t)1023;
    return p;
  };
  float* dinv = carve((size_t)N * 4);
  float* h1   = carve((size_t)N * F_HID * 4);
  float* out1 = carve((size_t)N * F_HID * 4);
  float* h2   = carve((size_t)N * 8 * 4);
  (void)ws_size; (void)n_in; (void)out_size;

  const int nb_node  = (N + 255) / 256;
  const int nb_edge  = (int)((E + 255) / 256);
  const int tiles    = (N + 15) / 16;            // 6250
  const int nb_gemm  = (tiles + 7) / 8;          // 8 waves (tiles) per block
  const int nb_n16   = (N * F_HID + 255) / 256;
  const int nb_e16   = (int)((E * F_HID + 255) / 256);
  const int nb_n8    = (N * 8 + 255) / 256;
  const int nb_e8    = (int)((E * 8 + 255) / 256);

  // Degree + symmetric normalization (shared by both layers).
  k_init_deg<<<nb_node, 256, 0, stream>>>(dinv, N);
  k_count_deg<<<nb_edge, 256, 0, stream>>>(ei, dinv, E);
  k_rsqrt_inplace<<<nb_node, 256, 0, stream>>>(dinv, N);

  // Layer 1: h1 = X @ W1 ; out1 = Agg(h1) + b1  (ReLU fused into GEMM2).
  k_gemm1_wmma<<<nb_gemm, 256, 0, stream>>>(X, W1, h1, N);
  k_agg1_init<<<nb_n16, 256, 0, stream>>>(h1, dinv, b1, out1, N);
  k_agg1_edge<<<nb_e16, 256, 0, stream>>>(ei, h1, dinv, out1, E);

  // Layer 2: h2 = relu(out1) @ W2 ; out = Agg(h2) + b2.
  k_gemm2_wmma<<<nb_gemm, 256, 0, stream>>>(out1, W2, h2, N);
  k_agg2_init<<<nb_n8, 256, 0, stream>>>(h2, dinv, b2, out, N);
  k_agg2_edge<<<nb_e8, 256, 0, stream>>>(ei, h2, dinv, out, E);
}